// Inference_75746043232775
// MI455X (gfx1250) — compile-verified
//
#include <hip/hip_runtime.h>
#include <stdint.h>

#define B_CNT 8
#define A_CNT 49104
#define C_CNT 90
#define IMG_F 512.0f
#define SCORE_THR 0.2f
#define IOU_THR 0.2f
#define K_PRE 1000
#define K_CAND 1024
#define K_OUT 100
#define TILE 128

// ---------------- CDNA5 async global->LDS copy path ----------------
// ISA: GLOBAL_LOAD_ASYNC_TO_LDS_B128 / _B32 (tracked on ASYNCcnt),
// completion via S_WAIT_ASYNCCNT. VDST = VGPR holding LDS byte address,
// VADDR = 64-bit global address (GV mode). Low 32 bits of a flat-cast
// __shared__ pointer are the wave-relative LDS byte offset (aperture
// decode truncates to addr[31:0]).
__device__ __forceinline__ void async_copy16(uint32_t lds_off, const void* g) {
  asm volatile("global_load_async_to_lds_b128 %0, %1, off"
               :: "v"(lds_off), "v"((uint64_t)(uintptr_t)g) : "memory");
}
__device__ __forceinline__ void async_copy4(uint32_t lds_off, const void* g) {
  asm volatile("global_load_async_to_lds_b32 %0, %1, off"
               :: "v"(lds_off), "v"((uint64_t)(uintptr_t)g) : "memory");
}
__device__ __forceinline__ void wait_async_zero() {
  asm volatile("s_wait_asynccnt 0" ::: "memory");
}

__device__ __forceinline__ unsigned fkey(float f) {
  unsigned u = __float_as_uint(f);
  return (u & 0x80000000u) ? ~u : (u | 0x80000000u); // monotone: bigger float -> bigger uint
}

// ---------------- Kernel 1: class max/argmax via async-LDS staging ----------------
__global__ __launch_bounds__(TILE) void k_score_class(
    const float* __restrict__ cls,
    float* __restrict__ scores, int* __restrict__ classes)
{
  __shared__ float smem[TILE * C_CNT]; // 46080 B, contiguous image of the tile
  const int b  = blockIdx.y;
  const int a0 = blockIdx.x * TILE;
  const int tid = threadIdx.x;
  int tile_n = A_CNT - a0; if (tile_n > TILE) tile_n = TILE;

  const float* gbase = cls + ((size_t)b * A_CNT + a0) * C_CNT;
  const unsigned bytes = (unsigned)tile_n * (C_CNT * 4u);
  const unsigned nch   = bytes >> 4;          // 16-byte chunks
  const unsigned remDW = (bytes & 15u) >> 2;  // leftover dwords
  const uint32_t lbase = (uint32_t)(uintptr_t)&smem[0];

  for (unsigned c = tid; c < nch; c += TILE)
    async_copy16(lbase + c * 16u, (const char*)gbase + (size_t)c * 16u);
  if ((unsigned)tid < remDW)
    async_copy4(lbase + nch * 16u + tid * 4u,
                (const char*)gbase + (size_t)nch * 16u + (size_t)tid * 4u);
  wait_async_zero();   // per-wave ASYNCcnt drain
  __syncthreads();     // cross-wave visibility of LDS

  if (tid < tile_n) {
    const float* row = &smem[tid * C_CNT];
    float mx = row[0]; int am = 0;
    #pragma unroll
    for (int c = 1; c < C_CNT; ++c) { float v = row[c]; if (v > mx) { mx = v; am = c; } }
    size_t gi = (size_t)b * A_CNT + (a0 + tid);
    scores[gi]  = (mx > SCORE_THR) ? mx : -1.0f;  // pre-masked like reference
    classes[gi] = am;
  }
}

// ---------------- Kernel 2: radix-select top-1000 + bitonic sort + decode ----------------
__global__ __launch_bounds__(1024) void k_select_sort(
    const float* __restrict__ scores, const int* __restrict__ classes,
    const float* __restrict__ loc, const float* __restrict__ anch,
    float* __restrict__ cand_score, int* __restrict__ cand_cls,
    float* __restrict__ cand_box, float* __restrict__ cand_obox)
{
  const int b = blockIdx.x;
  const int tid = threadIdx.x;
  const float* sc = scores + (size_t)b * A_CNT;

  __shared__ unsigned hist[256];
  __shared__ unsigned sh_prefix;
  __shared__ int      sh_remaining;
  __shared__ unsigned sh_cnt;
  __shared__ float s_val[K_CAND];
  __shared__ int   s_idx[K_CAND];

  if (tid == 0) { sh_prefix = 0u; sh_remaining = K_PRE; sh_cnt = 0u; }

  // 4x8-bit MSB-first radix select of the K_PRE-th largest key
  for (int pass = 0; pass < 4; ++pass) {
    const int shift = 24 - pass * 8;
    if (tid < 256) hist[tid] = 0u;
    __syncthreads();
    const unsigned prefix = sh_prefix;
    for (int a = tid; a < A_CNT; a += 1024) {
      unsigned k = fkey(sc[a]);
      unsigned hi = 0u;
      if (pass > 0) hi = k >> (shift + 8);
      if (hi == prefix) atomicAdd(&hist[(k >> shift) & 0xFFu], 1u);
    }
    __syncthreads();
    if (tid == 0) {
      unsigned p = sh_prefix; int rem = sh_remaining;
      for (int d = 255; d >= 0; --d) {
        int c = (int)hist[d];
        if (rem <= c) { p = (p << 8) | (unsigned)d; break; }
        rem -= c;
      }
      sh_prefix = p; sh_remaining = rem;
    }
    __syncthreads();
  }
  const unsigned T = sh_prefix;

  // compact: strictly greater first (guaranteed < K_PRE of them), then ties
  for (int a = tid; a < A_CNT; a += 1024) {
    if (fkey(sc[a]) > T) {
      unsigned p = atomicAdd(&sh_cnt, 1u);
      if (p < K_CAND) { s_val[p] = sc[a]; s_idx[p] = a; }
    }
  }
  __syncthreads();
  for (int a = tid; a < A_CNT; a += 1024) {
    if (fkey(sc[a]) == T) {
      unsigned p = atomicAdd(&sh_cnt, 1u);
      if (p < K_PRE) { s_val[p] = sc[a]; s_idx[p] = a; }
    }
  }
  __syncthreads();
  unsigned cnt = sh_cnt; if (cnt > K_PRE) cnt = K_PRE;
  if ((unsigned)tid >= cnt) { s_val[tid] = -2.0f; s_idx[tid] = 0; }
  __syncthreads();

  // bitonic sort, descending, 1024 elements, 1 thread per element
  for (unsigned k = 2; k <= K_CAND; k <<= 1) {
    for (unsigned j = k >> 1; j > 0; j >>= 1) {
      unsigned i = (unsigned)tid, ixj = i ^ j;
      if (ixj > i) {
        float vi = s_val[i], vj = s_val[ixj];
        bool descSeg = ((i & k) == 0);
        bool sw = descSeg ? (vi < vj) : (vi > vj);
        if (sw) {
          s_val[i] = vj; s_val[ixj] = vi;
          int t = s_idx[i]; s_idx[i] = s_idx[ixj]; s_idx[ixj] = t;
        }
      }
      __syncthreads();
    }
  }

  // gather + decode + clip only the 1024 candidates
  {
    const int r = tid;
    float v = s_val[r];
    int id = s_idx[r];
    size_t gi = (size_t)b * A_CNT + id;
    int c = classes[gi];
    float ay1 = anch[id*4+0], ax1 = anch[id*4+1], ay2 = anch[id*4+2], ax2 = anch[id*4+3];
    float ya = 0.5f*(ay1+ay2), xa = 0.5f*(ax1+ax2);
    float ha = ay2-ay1, wa = ax2-ax1;
    float dy = loc[gi*4+0], dx = loc[gi*4+1], dh = loc[gi*4+2], dw = loc[gi*4+3];
    float h = expf(dh)*ha, w = expf(dw)*wa;
    float yc = dy*ha + ya, xc = dx*wa + xa;
    float x1 = xc - 0.5f*w, y1 = yc - 0.5f*h, x2 = xc + 0.5f*w, y2 = yc + 0.5f*h;
    x1 = fminf(fmaxf(x1, 0.f), IMG_F); y1 = fminf(fmaxf(y1, 0.f), IMG_F);
    x2 = fminf(fmaxf(x2, 0.f), IMG_F); y2 = fminf(fmaxf(y2, 0.f), IMG_F);
    size_t o = (size_t)b * K_CAND + r;
    cand_score[o] = v;
    cand_cls[o]   = c;
    cand_box[o*4+0]=x1; cand_box[o*4+1]=y1; cand_box[o*4+2]=x2; cand_box[o*4+3]=y2;
    float off = (float)c * (IMG_F + 1.0f);  // batched-NMS class offset
    cand_obox[o*4+0]=x1+off; cand_obox[o*4+1]=y1+off; cand_obox[o*4+2]=x2+off; cand_obox[o*4+3]=y2+off;
  }
}

// ---------------- Kernel 3: suppression bitmask (transposed: word = j mod 32) ----------------
__global__ __launch_bounds__(256) void k_iou_mask(
    const float* __restrict__ cand_obox, unsigned* __restrict__ mask)
{
  __shared__ float sb[K_CAND * 4]; // 16KB: all candidate offset-boxes for this image
  const int b = blockIdx.y;
  const float* ob = cand_obox + (size_t)b * K_CAND * 4;
  for (int t = threadIdx.x; t < K_CAND * 4; t += 256) sb[t] = ob[t];
  __syncthreads();

  const int lane = threadIdx.x & 31;            // owns word 'lane': j = t*32 + lane
  const int ii = blockIdx.x * 8 + (threadIdx.x >> 5);
  float x1 = sb[ii*4+0], y1 = sb[ii*4+1], x2 = sb[ii*4+2], y2 = sb[ii*4+3];
  float ai = (x2 - x1) * (y2 - y1);
  unsigned m = 0u;
  #pragma unroll 4
  for (int t = 0; t < 32; ++t) {
    int j = t * 32 + lane;
    float a = sb[j*4+0], bb = sb[j*4+1], c = sb[j*4+2], d = sb[j*4+3];
    float xx1 = fmaxf(x1, a), yy1 = fmaxf(y1, bb);
    float xx2 = fminf(x2, c), yy2 = fminf(y2, d);
    float inter = fmaxf(xx2 - xx1, 0.f) * fmaxf(yy2 - yy1, 0.f);
    float aj = (c - a) * (d - bb);
    float iou = inter / (ai + aj - inter + 1e-8f);
    if (iou > IOU_THR) m |= (1u << t);
  }
  mask[((size_t)b * K_CAND + ii) * 32 + lane] = m;
}

// ---------------- Kernel 4: one wave32 per image, greedy NMS + emit ----------------
__global__ __launch_bounds__(32) void k_finalize(
    const float* __restrict__ cand_score, const int* __restrict__ cand_cls,
    const float* __restrict__ cand_box, const unsigned* __restrict__ mask,
    float* __restrict__ out)
{
  const int b = blockIdx.x;
  const int l = threadIdx.x;          // lane l owns removed bits for j == l (mod 32)
  unsigned removed = 0u;
  int cnt = 0;
  float* ob   = out + (size_t)b * K_OUT * 4;                                   // boxes
  float* osc  = out + (size_t)B_CNT * K_OUT * 4 + (size_t)b * K_OUT;           // scores
  float* olab = out + (size_t)B_CNT * K_OUT * 5 + (size_t)b * K_OUT;           // labels

  for (int i = 0; i < K_PRE; ++i) {
    unsigned rw = __shfl(removed, i & 31, 32);   // word holding candidate i's bit
    bool sup = (rw >> (i >> 5)) & 1u;
    float v = cand_score[(size_t)b * K_CAND + i];
    if (!sup && v > SCORE_THR) {
      removed |= mask[((size_t)b * K_CAND + i) * 32 + l];  // suppress everything i overlaps
      if (l == 0) {
        osc[cnt]  = v;
        olab[cnt] = (float)cand_cls[(size_t)b * K_CAND + i];
        const float* bx = cand_box + ((size_t)b * K_CAND + i) * 4;
        ob[cnt*4+0]=bx[0]; ob[cnt*4+1]=bx[1]; ob[cnt*4+2]=bx[2]; ob[cnt*4+3]=bx[3];
      }
      ++cnt;
      if (cnt >= K_OUT) break;
    }
  }
  for (int r = cnt + l; r < K_OUT; r += 32) {    // reference padding
    osc[r] = 0.f; olab[r] = -1.f;
    ob[r*4+0]=0.f; ob[r*4+1]=0.f; ob[r*4+2]=1.f; ob[r*4+3]=1.f;
  }
}

// ---------------- host ----------------
extern "C" void kernel_launch(void* const* d_in, const int* in_sizes, int n_in,
                              void* d_out, int out_size, void* d_ws, size_t ws_size,
                              hipStream_t stream) {
  (void)in_sizes; (void)n_in; (void)out_size; (void)ws_size;
  const float* cls  = (const float*)d_in[0];
  const float* loc  = (const float*)d_in[1];
  const float* anch = (const float*)d_in[2];
  float* out = (float*)d_out;
  char* ws = (char*)d_ws;

  size_t off = 0;
  auto take = [&](size_t bytes) { size_t o = off; off += (bytes + 255) & ~(size_t)255; return o; };
  float*    ws_scores  = (float*)   (ws + take((size_t)B_CNT * A_CNT * 4));
  int*      ws_cls     = (int*)     (ws + take((size_t)B_CNT * A_CNT * 4));
  float*    cand_score = (float*)   (ws + take((size_t)B_CNT * K_CAND * 4));
  int*      cand_cls   = (int*)     (ws + take((size_t)B_CNT * K_CAND * 4));
  float*    cand_box   = (float*)   (ws + take((size_t)B_CNT * K_CAND * 16));
  float*    cand_obox  = (float*)   (ws + take((size_t)B_CNT * K_CAND * 16));
  unsigned* mask       = (unsigned*)(ws + take((size_t)B_CNT * K_CAND * 32 * 4));

  dim3 g1((A_CNT + TILE - 1) / TILE, B_CNT);
  k_score_class<<<g1, TILE, 0, stream>>>(cls, ws_scores, ws_cls);
  k_select_sort<<<B_CNT, 1024, 0, stream>>>(ws_scores, ws_cls, loc, anch,
                                            cand_score, cand_cls, cand_box, cand_obox);
  k_iou_mask<<<dim3(K_CAND / 8, B_CNT), 256, 0, stream>>>(cand_obox, mask);
  k_finalize<<<B_CNT, 32, 0, stream>>>(cand_score, cand_cls, cand_box, mask, out);
}